// SingleCirculatePairConLoss_68152541053487
// MI455X (gfx1250) — compile-verified
//
#include <hip/hip_runtime.h>
#include <math.h>

// Problem constants (from reference)
#define N_SAMPLES 8192
#define D_FEAT    128
#define INV_T     20.0f                    // 1 / 0.05
#define TILE      16
#define WAVES     8
#define BLOCK_THREADS (WAVES * 32)         // 256
#define ROWS_PER_BLOCK (WAVES * TILE)      // 128 rows per block (1 row-tile per wave)
#define K_STEPS   (D_FEAT / 4)             // 32 wmma_f32_16x16x4 per 16x16 tile
#define NUM_CTILES (N_SAMPLES / TILE)      // 512 column tiles
#define CSPLIT    4                        // column split across blocks
#define TILES_PER_SPLIT (NUM_CTILES / CSPLIT)        // 128
#define NUM_ROWGROUPS (N_SAMPLES / ROWS_PER_BLOCK)   // 64
#define GRID_MAIN (NUM_ROWGROUPS * CSPLIT)           // 256 blocks
#define LDS_STRIDE 132                     // 128 + 4 pad floats -> conflict-free banks
#define BUF_FLOATS (TILE * LDS_STRIDE)     // 2112 floats per staging buffer
#define FINAL_THREADS 1024

typedef __attribute__((ext_vector_type(2))) float v2f;
typedef __attribute__((ext_vector_type(4))) float v4f;
typedef __attribute__((ext_vector_type(8))) float v8f;
typedef __attribute__((ext_vector_type(4))) int   v4i;

// ---- CDNA5 async global->LDS staging (ASYNCcnt path), with safe fallback ----
#if __has_builtin(__builtin_amdgcn_global_load_async_to_lds_b128)
#define HAVE_ASYNC 1
typedef __attribute__((address_space(1))) v4i as1_v4i;   // global int4
typedef __attribute__((address_space(3))) v4i as3_v4i;   // LDS int4
__device__ __forceinline__ void async_cp16(void* dst_lds, const void* src_glob) {
    __builtin_amdgcn_global_load_async_to_lds_b128(
        (as1_v4i*)src_glob,   // param0: int4 AS(1)* (per clang diagnostic)
        (as3_v4i*)dst_lds,    // param1: int4 AS(3)*
        0 /*imm offset*/, 0 /*cpol*/);
}
#else
#define HAVE_ASYNC 0
__device__ __forceinline__ void async_cp16(void* dst_lds, const void* src_glob) {
    *(v4f*)dst_lds = *(const v4f*)src_glob;   // synchronous fallback
}
#endif

template <int N>
__device__ __forceinline__ void wait_async_le() {
#if HAVE_ASYNC
#if __has_builtin(__builtin_amdgcn_s_wait_asynccnt)
    __builtin_amdgcn_s_wait_asynccnt(N);
#else
    asm volatile("s_wait_asynccnt %0" :: "i"(N) : "memory");
#endif
#endif
}

// Fused Gram-tile + shifted-exp reduction.
// Block = 8 waves; wave w owns rows [rowgroup*128 + 16w, +16). All waves share
// one async-staged, double-buffered LDS B tile; columns split 4-ways by block.
__global__ __launch_bounds__(BLOCK_THREADS, 1)
void paircon_main_kernel(const float* __restrict__ F,
                         const int*   __restrict__ lab,
                         float*       __restrict__ ws)
{
    __shared__ float smem[2 * BUF_FLOATS];

    const int tid      = threadIdx.x;
    const int lane     = tid & 31;
    const int wave     = tid >> 5;
    const int half     = lane >> 4;        // 0: lanes 0-15, 1: lanes 16-31
    const int l16      = lane & 15;
    const int rowgroup = blockIdx.x >> 2;
    const int split    = blockIdx.x & (CSPLIT - 1);
    const int rowbase  = rowgroup * ROWS_PER_BLOCK + wave * TILE;
    const int jt0      = split * TILES_PER_SPLIT;

    // A fragments (this wave's 16 rows x 128 K) resident in VGPRs.
    // ISA 16x4 f32 A layout: lane l16 holds row l16; pair = K {2*half, 2*half+1}.
    v2f afrag[K_STEPS];
    {
        const float* arow = F + (size_t)(rowbase + l16) * D_FEAT + 2 * half;
        #pragma unroll
        for (int s = 0; s < K_STEPS; ++s)
            afrag[s] = *(const v2f*)(arow + 4 * s);
    }

    // Per-row shift = ||f_row||^2 / T (diagonal logit), in registers.
    // Bitwise identical across column-split blocks -> partial sums add exactly.
    float nrm = 0.0f;
    #pragma unroll
    for (int s = 0; s < K_STEPS; ++s)
        nrm = fmaf(afrag[s].x, afrag[s].x, fmaf(afrag[s].y, afrag[s].y, nrm));
    nrm += __shfl_xor(nrm, 16, 32);        // lane l: ||f[rowbase+l16]||^2

    int   labr[8];
    float shiftr[8];
    #pragma unroll
    for (int r = 0; r < 8; ++r) {
        labr[r]   = lab[rowbase + 8 * half + r];
        shiftr[r] = __shfl(nrm, 8 * half + r, 32) * INV_T;
    }

    float rowsum[8], possum[8], cntf[8];
    #pragma unroll
    for (int r = 0; r < 8; ++r) { rowsum[r] = 0.0f; possum[r] = 0.0f; cntf[r] = 0.0f; }

    // Cooperative tile staging: 16 cols x 128 floats = 512 x b128 chunks,
    // 2 chunks per thread, written to padded row-major LDS layout.
    auto stage = [&](int buf, int jt) {
        #pragma unroll
        for (int i = 0; i < 2; ++i) {
            const int chunk = tid * 2 + i;     // 0..511
            const int col   = chunk >> 5;      // 0..15
            const int q     = chunk & 31;      // 0..31 (x 4 floats)
            const float* src = F + (size_t)(jt * TILE + col) * D_FEAT + q * 4;
            float*       dst = &smem[buf * BUF_FLOATS + col * LDS_STRIDE + q * 4];
            async_cp16(dst, src);
        }
    };

    stage(0, jt0);                              // prologue
    for (int t = 0; t < TILES_PER_SPLIT; ++t) {
        const int buf = t & 1;
        const int jt  = jt0 + t;

        // Issue next tile's DMA, then wait only for the current tile's 2 asyncs
        // (in-order completion) and make it visible block-wide.
        if (t + 1 < TILES_PER_SPLIT) { stage(buf ^ 1, jt + 1); wait_async_le<2>(); }
        else                         { wait_async_le<0>(); }
        __syncthreads();

        const int    labc  = lab[jt * TILE + l16];
        const float* bbase = &smem[buf * BUF_FLOATS + l16 * LDS_STRIDE + 2 * half];

        v8f c = {0.f, 0.f, 0.f, 0.f, 0.f, 0.f, 0.f, 0.f};
        #pragma unroll
        for (int s = 0; s < K_STEPS; ++s) {
            const v2f b = *(const v2f*)(bbase + 4 * s);   // ds_load_b64, bank-spread
            // D = A(16x4,f32) x B(4x16,f32) + C ; full fp32 matrix op.
            c = __builtin_amdgcn_wmma_f32_16x16x4_f32(
                    false, afrag[s], false, b, (short)0, c, false, false);
        }

        // c[r] = dot(row rowbase+8*half+r, col jt*16+l16). Shifted exp + sums.
        #pragma unroll
        for (int r = 0; r < 8; ++r) {
            const float e = __expf(fmaf(c[r], INV_T, -shiftr[r]));
            rowsum[r] += e;
            const bool match = (labc == labr[r]);
            possum[r] += match ? e : 0.0f;
            cntf[r]   += match ? 1.0f : 0.0f;
        }
        __syncthreads();    // everyone done with buf before it is re-staged
    }

    // Per-row (rs, ps, cf) partials: 16-lane shuffle reduce, one writer/row.
    #pragma unroll
    for (int r = 0; r < 8; ++r) {
        float rs = rowsum[r], ps = possum[r], cf = cntf[r];
        #pragma unroll
        for (int off = 8; off > 0; off >>= 1) {
            rs += __shfl_xor(rs, off, 32);
            ps += __shfl_xor(ps, off, 32);
            cf += __shfl_xor(cf, off, 32);
        }
        if (l16 == 0) {
            const int row = rowbase + 8 * half + r;
            const size_t o = ((size_t)row * CSPLIT + split) * 3;
            ws[o + 0] = rs;
            ws[o + 1] = ps;
            ws[o + 2] = cf;
        }
    }
}

// Combine the 4 column-split partials per row, take logs, and reduce all 8192
// per-row losses with a fixed-order tree (fully deterministic).
__global__ __launch_bounds__(FINAL_THREADS)
void paircon_finalize_kernel(const float* __restrict__ ws,
                             float* __restrict__ out)
{
    __shared__ float s[FINAL_THREADS];
    float acc = 0.0f;
    for (int row = threadIdx.x; row < N_SAMPLES; row += FINAL_THREADS) {
        float rs = 0.0f, ps = 0.0f, cf = 0.0f;
        #pragma unroll
        for (int k = 0; k < CSPLIT; ++k) {
            const size_t o = ((size_t)row * CSPLIT + k) * 3;
            rs += ws[o + 0];
            ps += ws[o + 1];
            cf += ws[o + 2];
        }
        // -log(pos/row) = log(row) - log(pos); shift cancels exactly.
        acc += (__logf(rs) - __logf(ps)) / cf;
    }
    s[threadIdx.x] = acc;
    __syncthreads();
    #pragma unroll
    for (int off = FINAL_THREADS / 2; off > 0; off >>= 1) {
        if ((int)threadIdx.x < off) s[threadIdx.x] += s[threadIdx.x + off];
        __syncthreads();
    }
    if (threadIdx.x == 0) out[0] = s[0];
}

extern "C" void kernel_launch(void* const* d_in, const int* in_sizes, int n_in,
                              void* d_out, int out_size, void* d_ws, size_t ws_size,
                              hipStream_t stream) {
    const float* F   = (const float*)d_in[0];   // f1: [8192,128] f32
    const int*   lab = (const int*)d_in[1];     // label: [8192] int
    float* out = (float*)d_out;                 // scalar loss
    float* ws  = (float*)d_ws;                  // 8192 rows x 4 splits x 3 floats

    paircon_main_kernel<<<GRID_MAIN, BLOCK_THREADS, 0, stream>>>(F, lab, ws);
    paircon_finalize_kernel<<<1, FINAL_THREADS, 0, stream>>>(ws, out);
}